// MultiScaleLeechQ_11020886081631
// MI455X (gfx1250) — compile-verified
//
#include <hip/hip_runtime.h>
#include <hip/hip_bf16.h>

// ---------------------------------------------------------------------------
// Multi-scale spherical residual VQ against the 196560-entry Leech codebook.
// Heavy op per scale: argmax_n ( tokens[N,24] . codebook[n,24] ) with
// V_WMMA_F32_16X16X4_F32 (fp32 WMMA, K=24 as 6 chained K=4 steps).
// 32-token M tile per wave: two accumulator chains share each B fragment
// (2x WMMA per codebook byte, independent chains hide WMMA RAW hazards).
// ---------------------------------------------------------------------------

#define N_EMBED    196560
#define NGROUPS    105          // 105 * 1872 == 196560, each group % 16 == 0
#define CHUNK      1872
#define NTOK_PAD   512          // max tokens per scale = 2*16*16
#define ZHAT_ELEMS 12288        // 2*24*1*16*16

typedef float v2f __attribute__((ext_vector_type(2)));
typedef float v8f __attribute__((ext_vector_type(8)));

__global__ void msq_init(const float* __restrict__ z, float* __restrict__ residual,
                         float* __restrict__ zhat, int n) {
  int i = blockIdx.x * blockDim.x + threadIdx.x;
  if (i < n) { residual[i] = z[i]; zhat[i] = 0.0f; }
}

// Anti-aliased separable triangle resize 16x16 -> h x w (jax.image.resize
// 'bilinear' downsample), output token-major [B*h*w, 24].
__global__ void msq_down(const float* __restrict__ res, float* __restrict__ tok,
                         int h, int w, int n) {
  int tid = blockIdx.x * blockDim.x + threadIdx.x;
  if (tid >= n) return;
  int d = tid % 24; int r = tid / 24;
  int x = r % w; r /= w;
  int y = r % h; int b = r / h;
  float sy = (float)h * (1.0f / 16.0f), sx = (float)w * (1.0f / 16.0f);
  float cy = (y + 0.5f) / sy - 0.5f;
  float cx = (x + 0.5f) / sx - 0.5f;
  float ky = fminf(sy, 1.0f), kx = fminf(sx, 1.0f);
  float wy[16], wx[16]; float sumy = 0.f, sumx = 0.f;
  for (int j = 0; j < 16; ++j) { float t = 1.0f - fabsf(((float)j - cy) * ky); wy[j] = fmaxf(t, 0.f); sumy += wy[j]; }
  for (int i = 0; i < 16; ++i) { float t = 1.0f - fabsf(((float)i - cx) * kx); wx[i] = fmaxf(t, 0.f); sumx += wx[i]; }
  const float* base = res + (b * 24 + d) * 256;
  float acc = 0.f;
  for (int j = 0; j < 16; ++j) {
    if (wy[j] == 0.f) continue;
    float rowacc = 0.f;
    for (int i = 0; i < 16; ++i) rowacc += wx[i] * base[j * 16 + i];
    acc += wy[j] * rowacc;
  }
  tok[((b * h + y) * w + x) * 24 + d] = acc / (sumy * sumx);
}

// One wave per (32-token tile, codebook group). fp32 WMMA 16x16x4, K=24.
// A layout (16x4 f32): lanes 0-15 -> M=lane, VGPR0=K0 VGPR1=K1; lanes 16-31 -> K2/K3.
// C/D layout: lane<16 -> N=lane, rows M=v (v=0..7); lane>=16 -> rows M=v+8.
__global__ void __launch_bounds__(32)
msq_argmax(const float* __restrict__ tok, const float* __restrict__ cb,
           int ntok, float* __restrict__ pvals, int* __restrict__ pidx) {
  const int lane = threadIdx.x;
  const int half = lane >> 4;          // 0 or 1
  const int l15  = lane & 15;
  const int m0   = blockIdx.x * 32;
  const int g    = blockIdx.y;

  int mA = m0 + l15;      if (mA >= ntok) mA = ntok - 1;   // pad by clamping
  int mB = m0 + 16 + l15; if (mB >= ntok) mB = ntok - 1;
  const float* ar0 = tok + mA * 24 + half * 2;
  const float* ar1 = tok + mB * 24 + half * 2;
  v2f a0[6], a1[6];
#pragma unroll
  for (int kc = 0; kc < 6; ++kc) {
    a0[kc].x = ar0[kc * 4]; a0[kc].y = ar0[kc * 4 + 1];
    a1[kc].x = ar1[kc * 4]; a1[kc].y = ar1[kc * 4 + 1];
  }

  float best0[8], best1[8]; int bi0[8], bi1[8];
#pragma unroll
  for (int v = 0; v < 8; ++v) {
    best0[v] = -3.0e38f; bi0[v] = 0;
    best1[v] = -3.0e38f; bi1[v] = 0;
  }

  const int e0 = g * CHUNK;
  const int e1 = e0 + CHUNK;           // CHUNK % 16 == 0, 105*CHUNK == N_EMBED
  // Per-lane codebook pointer; advances by a constant 16 rows * 96B per chunk.
  const float* brow = cb + (e0 + l15) * 24 + half * 2;
  for (int n0 = e0; n0 < e1; n0 += 16, brow += 16 * 24) {
    const int n = n0 + l15;            // this lane's codeword column
    if (n0 + 32 < e1)
      __builtin_prefetch(brow + 32 * 24, 0, 1);   // global_prefetch_b8, 2 chunks ahead
    v2f bf[6];
#pragma unroll
    for (int kc = 0; kc < 6; ++kc) { bf[kc].x = brow[kc * 4]; bf[kc].y = brow[kc * 4 + 1]; }
    v8f c0 = {}, c1 = {};
#pragma unroll
    for (int kc = 0; kc < 6; ++kc) {
#if __has_builtin(__builtin_amdgcn_wmma_f32_16x16x4_f32)
      c0 = __builtin_amdgcn_wmma_f32_16x16x4_f32(false, a0[kc], false, bf[kc],
                                                 (short)0, c0, false, false);
      c1 = __builtin_amdgcn_wmma_f32_16x16x4_f32(false, a1[kc], false, bf[kc],
                                                 (short)0, c1, false, false);
#else
      c0[0] += a0[kc].x * bf[kc].x + a0[kc].y * bf[kc].y;   // not expected on gfx1250
      c1[0] += a1[kc].x * bf[kc].x + a1[kc].y * bf[kc].y;
#endif
    }
#pragma unroll
    for (int v = 0; v < 8; ++v) {
      if (c0[v] > best0[v]) { best0[v] = c0[v]; bi0[v] = n; }  // ascending n: first max wins
      if (c1[v] > best1[v]) { best1[v] = c1[v]; bi1[v] = n; }
    }
  }

  // argmax over the 16 N-columns: xor masks 1..8 stay inside each 16-lane half
#pragma unroll
  for (int off = 1; off < 16; off <<= 1) {
#pragma unroll
    for (int v = 0; v < 8; ++v) {
      float ov = __shfl_xor(best0[v], off, 32);
      int   oi = __shfl_xor(bi0[v], off, 32);
      if (ov > best0[v] || (ov == best0[v] && oi < bi0[v])) { best0[v] = ov; bi0[v] = oi; }
      float pv = __shfl_xor(best1[v], off, 32);
      int   pi = __shfl_xor(bi1[v], off, 32);
      if (pv > best1[v] || (pv == best1[v] && pi < bi1[v])) { best1[v] = pv; bi1[v] = pi; }
    }
  }
  if (l15 == 0) {
#pragma unroll
    for (int v = 0; v < 8; ++v) {
      int t0 = m0 + v + half * 8;
      if (t0 < ntok) { pvals[g * NTOK_PAD + t0] = best0[v]; pidx[g * NTOK_PAD + t0] = bi0[v]; }
      int t1 = m0 + 16 + v + half * 8;
      if (t1 < ntok) { pvals[g * NTOK_PAD + t1] = best1[v]; pidx[g * NTOK_PAD + t1] = bi1[v]; }
    }
  }
}

__global__ void msq_reduce(const float* __restrict__ pvals, const int* __restrict__ pidx,
                           int ntok, int* __restrict__ idx_out) {
  int t = blockIdx.x * blockDim.x + threadIdx.x;
  if (t >= ntok) return;
  float best = -3.0e38f; int bi = 0x7fffffff;
  for (int g = 0; g < NGROUPS; ++g) {
    float v = pvals[g * NTOK_PAD + t];
    int   i = pidx [g * NTOK_PAD + t];
    if (v > best || (v == best && i < bi)) { best = v; bi = i; }
  }
  idx_out[t] = bi;
}

// Gather codebook[idx], bilinear-upsample h x w -> 16x16, z_hat += / residual -=.
__global__ void msq_accum(const float* __restrict__ cb, const int* __restrict__ idxbuf,
                          float* __restrict__ residual, float* __restrict__ zhat,
                          int h, int w) {
  int tid = blockIdx.x * blockDim.x + threadIdx.x;
  if (tid >= ZHAT_ELEMS) return;
  int X = tid & 15; int Y = (tid >> 4) & 15; int d = (tid >> 8) % 24; int b = tid / 6144;
  float cy = (Y + 0.5f) * (float)h * (1.0f / 16.0f) - 0.5f;   // upsample: radius-1 kernel
  float cx = (X + 0.5f) * (float)w * (1.0f / 16.0f) - 0.5f;
  float wy[16], wx[16]; float sumy = 0.f, sumx = 0.f;
  for (int j = 0; j < h; ++j) { float t = 1.0f - fabsf((float)j - cy); wy[j] = fmaxf(t, 0.f); sumy += wy[j]; }
  for (int i = 0; i < w; ++i) { float t = 1.0f - fabsf((float)i - cx); wx[i] = fmaxf(t, 0.f); sumx += wx[i]; }
  float acc = 0.f;
  for (int j = 0; j < h; ++j) {
    if (wy[j] == 0.f) continue;
    float rowacc = 0.f;
    for (int i = 0; i < w; ++i) {
      if (wx[i] == 0.f) continue;
      int code = idxbuf[(b * h + j) * w + i];
      rowacc += wx[i] * cb[code * 24 + d];   // codebook rows are unit-norm
    }
    acc += wy[j] * rowacc;
  }
  acc /= (sumy * sumx);
  zhat[tid]     += acc;
  residual[tid] -= acc;
}

extern "C" void kernel_launch(void* const* d_in, const int* in_sizes, int n_in,
                              void* d_out, int out_size, void* d_ws, size_t ws_size,
                              hipStream_t stream) {
  (void)in_sizes; (void)n_in; (void)out_size; (void)ws_size;
  const float* z  = (const float*)d_in[0];   // [2,24,1,16,16]
  const float* cb = (const float*)d_in[1];   // [196560,24], unit rows

  float* zhat     = (float*)d_out;                        // 12288 floats
  int*   idx_out  = (int*)(zhat + ZHAT_ELEMS);            // 1360 int32 codes

  float* residual = (float*)d_ws;                         // 12288
  float* tok      = residual + ZHAT_ELEMS;                // up to 512*24
  float* pvals    = tok + ZHAT_ELEMS;                     // NGROUPS*NTOK_PAD
  int*   pidx     = (int*)(pvals + NGROUPS * NTOK_PAD);   // NGROUPS*NTOK_PAD

  msq_init<<<(ZHAT_ELEMS + 255) / 256, 256, 0, stream>>>(z, residual, zhat, ZHAT_ELEMS);

  static const int HW[10] = {1, 2, 3, 4, 5, 6, 8, 10, 13, 16};
  int off = 0;
  for (int s = 0; s < 10; ++s) {
    const int h = HW[s], w = HW[s];
    const int ntok = 2 * h * w;
    const int ndown = ntok * 24;

    msq_down<<<(ndown + 255) / 256, 256, 0, stream>>>(residual, tok, h, w, ndown);

    dim3 grid((ntok + 31) / 32, NGROUPS);
    msq_argmax<<<grid, 32, 0, stream>>>(tok, cb, ntok, pvals, pidx);

    msq_reduce<<<(ntok + 63) / 64, 64, 0, stream>>>(pvals, pidx, ntok, idx_out + off);

    msq_accum<<<(ZHAT_ELEMS + 255) / 256, 256, 0, stream>>>(cb, idx_out + off,
                                                            residual, zhat, h, w);
    off += ntok;
  }
}